// QueryAndGroup_52785148067965
// MI455X (gfx1250) — compile-verified
//
#include <hip/hip_runtime.h>
#include <hip/hip_bf16.h>
#include <stddef.h>

typedef __attribute__((ext_vector_type(2))) float v2f;
typedef __attribute__((ext_vector_type(8))) float v8f;

#define BATCH    16
#define NPTS     4096
#define NCTR     1024
#define NCH      64
#define NSAMPLE  32
#define RADIUS2  0.04f
#define CH_OUT   (3 + NCH)     // 67
#define WAVES_PB 4             // waves per block in ball-query kernel

// ---------------------------------------------------------------------------
// Kernel A: ball query via V_WMMA_F32_16X16X4_F32, software-pipelined.
// One wave handles a tile of 16 centers; loops over N in chunks of 16 points.
// d2[m][n] = |q_m|^2 - 2 q_m.p_n + |p_n|^2 computed as a K=4 WMMA:
//   A[m][0..3] = (qx, qy, qz, |q|^2)   B[0..3][n] = (-2px, -2py, -2pz, 1)
//   C[m][n]    = |p_n|^2  (depends only on n -> same value in all 8 acc VGPRs)
// A layout (32-bit 16x4): lanes 0-15 hold K=0 (v0), K=1 (v1); lanes 16-31 hold
// K=2 (v0), K=3 (v1). B layout mirrored. D layout: vgpr r, lanes 0-15 -> M=r,
// lanes 16-31 -> M=r+8.
// The next tile's coordinates are loaded BEFORE the current tile is processed,
// so the WMMA + ballot + append work hides the load latency (only ~1 wave per
// SIMD is resident in this kernel, so latency must be hidden in-wave).
// ---------------------------------------------------------------------------
__global__ void __launch_bounds__(WAVES_PB * 32)
ballquery_wmma_kernel(const float* __restrict__ xyz,
                      const float* __restrict__ new_xyz,
                      int* __restrict__ idx)
{
    const int wave  = threadIdx.x >> 5;
    const int lane  = threadIdx.x & 31;
    const int tile  = blockIdx.x * WAVES_PB + wave;   // 0 .. B*NCTR/16 - 1
    const int b     = tile >> 6;                      // NCTR/16 = 64 tiles per batch
    const int m0    = (tile & 63) << 4;               // first center of tile
    const int mlane = lane & 15;
    const bool hi   = (lane >= 16);

    // ---- A matrix: 16 centers, K=4 ----
    const float* qc = new_xyz + ((size_t)b * NCTR + (m0 + mlane)) * 3;
    const float qx = qc[0], qy = qc[1], qz = qc[2];
    const float qn2 = qx * qx + qy * qy + qz * qz;
    v2f Amat;
    Amat.x = hi ? qz  : qx;
    Amat.y = hi ? qn2 : qy;

    // ---- per-lane selection state: lane t (t<16) owns center m0+t ----
    int cnt   = hi ? NSAMPLE : 0;   // upper lanes are permanently "done"
    int first = 0;
    int* myidx = idx + ((size_t)b * NCTR + (m0 + mlane)) * NSAMPLE;

    const float* xbase = xyz + (size_t)b * NPTS * 3;

    // prologue: load tile 0 coordinates
    float px = xbase[mlane * 3 + 0];
    float py = xbase[mlane * 3 + 1];
    float pz = xbase[mlane * 3 + 2];

    for (int nb = 0; nb < NPTS; nb += 16) {
        // ---- issue next tile's loads first (double buffer) ----
        float nx = 0.0f, ny = 0.0f, nz = 0.0f;
        if (nb + 16 < NPTS) {
            const float* pn = xbase + (size_t)(nb + 16 + mlane) * 3;
            nx = pn[0]; ny = pn[1]; nz = pn[2];
        }
        // long-range prefetch (global_prefetch_b8) to pre-warm the caches
        if (nb + 128 < NPTS) {
            __builtin_prefetch(xbase + (size_t)(nb + 128 + mlane) * 3, 0, 0);
        }

        // ---- B matrix + C (|p|^2) for the current 16 candidate points ----
        const float pn2 = px * px + py * py + pz * pz;
        v2f Bmat;
        Bmat.x = hi ? (-2.0f * pz) : (-2.0f * px);
        Bmat.y = hi ? 1.0f         : (-2.0f * py);
        v8f Cmat;
#pragma unroll
        for (int r = 0; r < 8; ++r) Cmat[r] = pn2;

        // D[m][n] = squared distance (uniform control flow -> EXEC all 1s)
        v8f D = __builtin_amdgcn_wmma_f32_16x16x4_f32(
            false, Amat, false, Bmat, (short)0, Cmat, false, false);

        // ---- 16x16 compare results -> per-center 16-bit masks via ballot ----
        unsigned sub = 0;
#pragma unroll
        for (int r = 0; r < 8; ++r) {
            unsigned bal = (unsigned)__ballot(D[r] < RADIUS2);
            sub = (mlane == r)     ? (bal & 0xFFFFu) : sub;   // center r
            sub = (mlane == r + 8) ? (bal >> 16)     : sub;   // center r+8
        }
        if (hi) sub = 0;

        // ---- append in index order until 32 filled ----
        while (sub && cnt < NSAMPLE) {
            const int bit = __builtin_ctz(sub);
            sub &= sub - 1;
            const int n = nb + bit;
            if (cnt == 0) first = n;
            myidx[cnt] = n;
            ++cnt;
        }

        // rotate the pipeline registers
        px = nx; py = ny; pz = nz;

        // wave-uniform early exit once all 16 centers are full
        if (__ballot(cnt < NSAMPLE) == 0) break;
    }

    // unfilled slots repeat the first in-radius index (0 if none)
    if (!hi) {
        for (int s = cnt; s < NSAMPLE; ++s) myidx[s] = first;
    }
}

// ---------------------------------------------------------------------------
// Kernel B: gather/group. One block per (np, b); idx + center in LDS.
// 8 waves stride over the 67 output channels; lane s handles sample s so
// every store is a fully coalesced 128B row. points (16.7 MB) lives in L2;
// output (140 MB) is written once with non-temporal stores so the streaming
// writes don't evict the gather working set from L2.
// out layout: (B, 67, NP, 32); channels 0..2 are grouped_xyz - center.
// ---------------------------------------------------------------------------
__global__ void __launch_bounds__(256)
group_kernel(const float* __restrict__ xyz,
             const float* __restrict__ new_xyz,
             const float* __restrict__ points,
             const int* __restrict__ idx,
             float* __restrict__ out)
{
    const int np = blockIdx.x;   // 0..1023
    const int b  = blockIdx.y;   // 0..15

    __shared__ int   sidx[NSAMPLE];
    __shared__ float scen[3];
    if (threadIdx.x < NSAMPLE)
        sidx[threadIdx.x] = idx[((size_t)b * NCTR + np) * NSAMPLE + threadIdx.x];
    if (threadIdx.x < 3)
        scen[threadIdx.x] = new_xyz[((size_t)b * NCTR + np) * 3 + threadIdx.x];
    __syncthreads();

    const int lane = threadIdx.x & 31;
    const int w    = threadIdx.x >> 5;    // 0..7
    const int n    = sidx[lane];

    const float* pb = points + (size_t)b * NCH * NPTS;
    const float* xb = xyz    + (size_t)b * NPTS * 3;

    for (int c = w; c < CH_OUT; c += 8) {
        float v;
        if (c < 3) v = xb[n * 3 + c] - scen[c];
        else       v = pb[(size_t)(c - 3) * NPTS + n];
        float* dst = out + (((size_t)b * CH_OUT + c) * NCTR + np) * NSAMPLE + lane;
        __builtin_nontemporal_store(v, dst);
    }
}

// ---------------------------------------------------------------------------
extern "C" void kernel_launch(void* const* d_in, const int* in_sizes, int n_in,
                              void* d_out, int out_size, void* d_ws, size_t ws_size,
                              hipStream_t stream)
{
    const float* xyz     = (const float*)d_in[0];  // (B, N, 3)
    const float* new_xyz = (const float*)d_in[1];  // (B, NP, 3)
    const float* points  = (const float*)d_in[2];  // (B, C, N)
    float* out = (float*)d_out;                    // (B, 67, NP, 32)
    int*   idx = (int*)d_ws;                       // (B, NP, 32) = 2 MB scratch

    // Kernel A: B*NCTR/16 = 1024 center tiles, WAVES_PB tiles per block.
    const int n_tiles = BATCH * NCTR / 16;
    ballquery_wmma_kernel<<<n_tiles / WAVES_PB, WAVES_PB * 32, 0, stream>>>(
        xyz, new_xyz, idx);

    // Kernel B: one block per (np, b) = 16384 blocks x 256 threads.
    dim3 grid(NCTR, BATCH);
    group_kernel<<<grid, 256, 0, stream>>>(xyz, new_xyz, points, idx, out);
}